// LightGCNLayer_24739011625577
// MI455X (gfx1250) — compile-verified
//
#include <hip/hip_runtime.h>
#include <stdint.h>

// LightGCN layer: degree-normalized gather-scatter.
// d = 32 features == wave32 lanes: one wave per edge.
#define FEAT_D 32

// ---------------------------------------------------------------------------
// CDNA5 async global->LDS copy (ASYNCcnt-tracked) with inline-asm fallback.
// Builtin takes typed v2i32 addrspace(1)/(3) pointers (per hipcc diagnostic).
// ---------------------------------------------------------------------------
#if defined(__has_builtin)
#  if __has_builtin(__builtin_amdgcn_global_load_async_to_lds_b64)
#    define HAVE_ASYNC_LDS_BUILTIN 1
#  endif
#  if __has_builtin(__builtin_amdgcn_s_wait_asynccnt)
#    define HAVE_WAIT_ASYNC_BUILTIN 1
#  endif
#endif

typedef int v2i32 __attribute__((__vector_size__(8)));

__device__ __forceinline__ void async_b64_to_lds(const long long* gptr,
                                                 long long* lptr) {
#if defined(HAVE_ASYNC_LDS_BUILTIN)
  __builtin_amdgcn_global_load_async_to_lds_b64(
      (__attribute__((address_space(1))) v2i32*)(void*)gptr,
      (__attribute__((address_space(3))) v2i32*)(void*)lptr,
      /*offset=*/0, /*cpol=*/0);
#else
  unsigned lds_addr =
      (unsigned)(uintptr_t)(__attribute__((address_space(3))) char*)(void*)lptr;
  asm volatile("global_load_async_to_lds_b64 %0, %1, off"
               :
               : "v"(lds_addr), "v"(gptr)
               : "memory");
#endif
}

// Wait until ASYNCcnt <= N (N must be a compile-time constant).
template <int N>
__device__ __forceinline__ void wait_async_le() {
#if defined(HAVE_WAIT_ASYNC_BUILTIN)
  __builtin_amdgcn_s_wait_asynccnt(N);
#else
  if (N == 0)
    asm volatile("s_wait_asynccnt 0x0" ::: "memory");
  else
    asm volatile("s_wait_asynccnt 0x2" ::: "memory");
#endif
  asm volatile("" ::: "memory");  // keep LDS reads below the wait
}

// ---------------------------------------------------------------------------
// Edge scatter (FIRST kernel in file so the disasm snippet shows the async
// path). One wave per 32-edge tile, double-buffered async index staging:
//   buf A processing overlaps buf B prefetch; s_wait_asynccnt<=2 releases the
//   current buffer (async loads complete in order) while the next tile's two
//   loads stay in flight.
// Per edge: lane-coalesced 128B row gather, scale by norm_src, f32 global
// atomic add into the L2-resident (12.8 MB << 192 MB) output row.
// ---------------------------------------------------------------------------
__global__ __launch_bounds__(256) void k_scatter(
    const float* __restrict__ h_src, const long long* __restrict__ es,
    const long long* __restrict__ ed, const float* __restrict__ norm_src,
    float* __restrict__ out, long long n_edges) {
  // 8 waves * 2 buffers * (32 src + 32 dst) int64 = 8 KB LDS
  __shared__ long long s_edges[8 * 2 * 64];

  const int lane = threadIdx.x & 31;
  const int wid = threadIdx.x >> 5;
  long long* slot0 = &s_edges[wid * 128];
  long long* slot1 = slot0 + 64;

  const long long n_tiles = (n_edges + 31) >> 5;
  const long long wave_stride = (long long)gridDim.x * 8;
  long long t = (long long)blockIdx.x * 8 + wid;

  // Prologue: stage first tile (if it is a full 32-edge tile).
  bool cur_async = false;
  long long* cur = slot0;
  long long* nxt = slot1;
  if (t < n_tiles && (n_edges - (t << 5)) >= 32) {
    async_b64_to_lds(es + (t << 5) + lane, cur + lane);
    async_b64_to_lds(ed + (t << 5) + lane, cur + 32 + lane);
    cur_async = true;
  }

  while (t < n_tiles) {
    const long long tn = t + wave_stride;
    // Prefetch next tile into the alternate buffer.
    bool nxt_async = false;
    if (tn < n_tiles && (n_edges - (tn << 5)) >= 32) {
      async_b64_to_lds(es + (tn << 5) + lane, nxt + lane);
      async_b64_to_lds(ed + (tn << 5) + lane, nxt + 32 + lane);
      nxt_async = true;
    }

    const long long e0 = t << 5;
    if (cur_async) {
      if (nxt_async)
        wait_async_le<2>();  // current buffer done; prefetch still in flight
      else
        wait_async_le<0>();
#pragma unroll 4
      for (int k = 0; k < 32; ++k) {
        const int s = (int)cur[k];       // LDS broadcast (conflict-free)
        const int d = (int)cur[32 + k];
        const float v = h_src[s * FEAT_D + lane] * norm_src[s];
        atomicAdd(&out[d * FEAT_D + lane], v);
      }
    } else {
      // Tail tile (not hit for 1.6M edges; kept for generality).
      const int cnt = (int)(n_edges - e0);
      for (int k = 0; k < cnt; ++k) {
        const int s = (int)es[e0 + k];
        const int d = (int)ed[e0 + k];
        const float v = h_src[s * FEAT_D + lane] * norm_src[s];
        atomicAdd(&out[d * FEAT_D + lane], v);
      }
    }

    long long* tmp = cur; cur = nxt; nxt = tmp;
    cur_async = nxt_async;
    t = tn;
  }
}

// ---------------------------------------------------------------------------
// Zero a u32 region (degree scratch and f32 output; 0 bits == 0.0f).
// Harness poisons d_out/d_ws, so we must zero every call.
// ---------------------------------------------------------------------------
__global__ void k_zero_u32(unsigned* __restrict__ p, long long n) {
  long long i = (long long)blockIdx.x * blockDim.x + threadIdx.x;
  long long stride = (long long)gridDim.x * blockDim.x;
  for (; i < n; i += stride) p[i] = 0u;
}

// ---------------------------------------------------------------------------
// Degree histogram. 3.2M u32 atomics into 800 KB (L2-resident).
// ---------------------------------------------------------------------------
__global__ void k_degrees(const long long* __restrict__ es,
                          const long long* __restrict__ ed,
                          unsigned* __restrict__ deg_src,
                          unsigned* __restrict__ deg_dst, long long n_edges) {
  long long i = (long long)blockIdx.x * blockDim.x + threadIdx.x;
  long long stride = (long long)gridDim.x * blockDim.x;
  for (; i < n_edges; i += stride) {
    atomicAdd(&deg_src[(int)es[i]], 1u);
    atomicAdd(&deg_dst[(int)ed[i]], 1u);
  }
}

// ---------------------------------------------------------------------------
// In-place deg(u32) -> norm(f32) = 1/sqrt(max(deg,1)); covers both contiguous
// degree arrays in one pass.
// ---------------------------------------------------------------------------
__global__ void k_norms(unsigned* __restrict__ buf, long long n) {
  long long i = (long long)blockIdx.x * blockDim.x + threadIdx.x;
  long long stride = (long long)gridDim.x * blockDim.x;
  float* fbuf = (float*)buf;
  for (; i < n; i += stride) {
    unsigned d = buf[i];
    float dd = (float)(d < 1u ? 1u : d);
    fbuf[i] = 1.0f / sqrtf(dd);
  }
}

// ---------------------------------------------------------------------------
// out[v, f] *= norm_dst[v]
// ---------------------------------------------------------------------------
__global__ void k_scale(float* __restrict__ out,
                        const float* __restrict__ norm_dst, long long n) {
  long long i = (long long)blockIdx.x * blockDim.x + threadIdx.x;
  long long stride = (long long)gridDim.x * blockDim.x;
  for (; i < n; i += stride) out[i] *= norm_dst[i >> 5];  // i / FEAT_D
}

// ---------------------------------------------------------------------------
static inline int blocks_for(long long n, int cap) {
  long long b = (n + 255) / 256;
  if (b < 1) b = 1;
  if (b > cap) b = cap;
  return (int)b;
}

extern "C" void kernel_launch(void* const* d_in, const int* in_sizes, int n_in,
                              void* d_out, int out_size, void* d_ws,
                              size_t ws_size, hipStream_t stream) {
  const float* h_src = (const float*)d_in[0];
  // d_in[1] (h_dst) only contributes its shape in the reference; unused here.
  const long long* es = (const long long*)d_in[2];
  const long long* ed = (const long long*)d_in[3];
  float* out = (float*)d_out;

  const long long n_src = (long long)in_sizes[0] / FEAT_D;
  const long long n_dst = (long long)in_sizes[1] / FEAT_D;
  const long long n_edges = (long long)in_sizes[2];

  // Workspace layout: [deg/norm_src : n_src u32][deg/norm_dst : n_dst u32]
  unsigned* deg_src = (unsigned*)d_ws;
  unsigned* deg_dst = deg_src + n_src;
  const long long n_deg = n_src + n_dst;

  // 1) zero degree scratch and output accumulator
  k_zero_u32<<<blocks_for(n_deg, 2048), 256, 0, stream>>>(deg_src, n_deg);
  k_zero_u32<<<blocks_for(out_size, 4096), 256, 0, stream>>>((unsigned*)out,
                                                             (long long)out_size);
  // 2) degrees
  k_degrees<<<blocks_for(n_edges, 4096), 256, 0, stream>>>(es, ed, deg_src,
                                                           deg_dst, n_edges);
  // 3) degrees -> inverse-sqrt norms (in place, both arrays contiguous)
  k_norms<<<blocks_for(n_deg, 2048), 256, 0, stream>>>(deg_src, n_deg);
  // 4) gather-scale-scatter over edges (one wave per 32-edge tile,
  //    double-buffered async index staging)
  {
    const long long n_tiles = (n_edges + 31) >> 5;
    const int blocks = blocks_for(n_tiles * 32, 4096);  // 8 waves/block
    k_scatter<<<blocks, 256, 0, stream>>>(h_src, es, ed, (const float*)deg_src,
                                          out, n_edges);
  }
  // 5) scale rows by norm_dst
  k_scale<<<blocks_for(out_size, 4096), 256, 0, stream>>>(
      out, (const float*)deg_dst, (long long)out_size);
}